// ObjectDetectionHead_10067403342313
// MI455X (gfx1250) — compile-verified
//
#include <hip/hip_runtime.h>
#include <math.h>

#define BATCH 2
#define CIN   64
#define DD    64
#define HH    64
#define WW    64
#define LL    (DD*HH*WW)       // 262144 anchors per batch
#define NCLS  5
#define NCH   8                // 5 cls + 3 off, padded to 16 in the WMMA N dim
#define NMAX  32
#define TOPKK 13
#define NSLOT (NMAX*TOPKK)     // 416
#define EPSF  1e-9f
#define TAPS  27
#define CHALF 32               // channels per TDM plane load

typedef float v2f __attribute__((ext_vector_type(2)));
typedef float v8f __attribute__((ext_vector_type(8)));
typedef unsigned int u32x4 __attribute__((ext_vector_type(4)));
typedef int i32x4 __attribute__((ext_vector_type(4)));
typedef int i32x8 __attribute__((ext_vector_type(8)));

__device__ __forceinline__ float sigmoidf_(float x) { return 1.0f / (1.0f + expf(-x)); }
__device__ __forceinline__ float softplusf_(float x) { return log1pf(expf(-fabsf(x))) + fmaxf(x, 0.0f); }

// ---------------------------------------------------------------------------
// TDM: DMA a 2-D tile (CHALF channel-rows x 64 w, row stride = LL elements)
// from global memory into LDS. Descriptor per CDNA5 ISA ch.8 (D# groups).
// ---------------------------------------------------------------------------
__device__ __forceinline__ void tdm_load_plane(const float* gsrc, unsigned lds_byte_off)
{
  const unsigned long long ga = (unsigned long long)(uintptr_t)gsrc;
  u32x4 g0;
  g0[0] = 1u;                                            // count=1 (valid D#)
  g0[1] = lds_byte_off;                                  // lds_addr (bytes)
  g0[2] = (unsigned)(ga & 0xffffffffu);                  // global_addr[31:0]
  g0[3] = (unsigned)((ga >> 32) & 0x01ffffffu) | (2u << 30); // addr[56:32] | type=2
  i32x8 g1;
  g1[0] = (int)(2u << 16);                               // data_size = 4 bytes
  g1[1] = (int)(64u << 16);                              // tensor_dim0 = 64 (w)
  g1[2] = (int)(64u << 16);                              // tensor_dim1 = 64 (c rows)
  g1[3] = (int)(64u << 16);                              // tile_dim0 = 64
  g1[4] = (int)CHALF;                                    // tile_dim1 = 32 rows
  g1[5] = (int)(unsigned)LL;                             // tensor_dim0_stride lo
  g1[6] = 0;                                             // stride hi / dim1_stride lo
  g1[7] = 0;                                             // dim1_stride hi
  const i32x4 gz = {0, 0, 0, 0};                         // 2-D: groups 2/3 unused
#if __clang_major__ >= 23
  const i32x8 gz8 = {0, 0, 0, 0, 0, 0, 0, 0};
  __builtin_amdgcn_tensor_load_to_lds(g0, g1, gz, gz, gz8, 0);
#else
  __builtin_amdgcn_tensor_load_to_lds(g0, g1, gz, gz, 0);
#endif
}

// ---------------------------------------------------------------------------
// K1: implicit-GEMM 3x3x3 conv via V_WMMA_F32_16X16X4_F32, fed by TDM.
// Block = one (b,d,h) row of 64 w; wave = one 16-wide w-tile. For each of the
// 9 (kd,kh) planes, the TDM stages 32-channel x 64-w slabs into LDS (zeroed
// for SAME-padding OOB planes); all 4 waves consume them for 3 kw taps.
// ---------------------------------------------------------------------------
__global__ __launch_bounds__(128) void conv_wmma_kernel(
    const float* __restrict__ feat,
    const float* __restrict__ cls_w, const float* __restrict__ cls_b,
    const float* __restrict__ off_w, const float* __restrict__ off_b,
    float* __restrict__ logits, float* __restrict__ centers)
{
  __shared__ float wlds[TAPS * CIN * NCH];   // 13824 floats = 55296 B
  __shared__ float plane[CHALF * WW];        //  2048 floats =  8192 B
  for (int i = threadIdx.x; i < TAPS * CIN * NCH; i += blockDim.x) {
    const int n = i & 7, c = (i >> 3) & 63, tap = i >> 9;
    float v;
    if (n < NCLS) v = cls_w[(n * CIN + c) * TAPS + tap];
    else          v = off_w[((n - NCLS) * CIN + c) * TAPS + tap];
    wlds[i] = v;
  }

  const int lane = threadIdx.x & 31;
  const int wave = threadIdx.x >> 5;
  const int tile = blockIdx.x * 4 + wave;        // 4 waves = 4 w-tiles, same (b,d,h)
  const int w0 = (tile & 3) << 4;
  const int hy = (tile >> 2) & 63;               // uniform across the block
  const int dz = (tile >> 8) & 63;               // uniform across the block
  const int b  = tile >> 14;                     // uniform across the block

  const int half = lane >> 4;                    // K-pair select (ISA A/B layout)
  const int klo  = half << 1;
  const int mn   = lane & 15;                    // A row m == B col n
  const int wl   = w0 + mn;

  const unsigned plane_off = (unsigned)(uintptr_t)plane;   // LDS byte offset
  const size_t bbase = (size_t)b * CIN * LL;
  v8f acc = {};

  for (int kd = 0; kd < 3; ++kd) {
    const int d2i = dz + kd - 1;
    for (int kh = 0; kh < 3; ++kh) {
      const int h2i = hy + kh - 1;
      const bool pok = ((unsigned)d2i < DD) && ((unsigned)h2i < HH);
      for (int chalf = 0; chalf < CIN; chalf += CHALF) {
        __syncthreads();                         // previous plane fully consumed
        if (pok) {
          if (wave == 0) {
            const float* gsrc = feat + bbase + (size_t)chalf * LL
                              + ((size_t)d2i * HH + h2i) * WW;
            tdm_load_plane(gsrc, plane_off);
            __builtin_amdgcn_s_wait_tensorcnt((short)0);
          }
        } else {
          for (int i = threadIdx.x; i < CHALF * WW; i += 128) plane[i] = 0.0f;
        }
        __syncthreads();                         // plane published to all waves

        for (int kw = 0; kw < 3; ++kw) {
          const int w2i = wl + kw - 1;
          const bool wok = (unsigned)w2i < WW;   // SAME padding in w (per lane)
          const int tap = (kd * 3 + kh) * 3 + kw;
          const float* wbase = wlds + tap * CIN * NCH;
          for (int ck = 0; ck < CHALF; ck += 4) {
            const int cl = ck + klo;             // channel row within plane
            const int cg = chalf + cl;           // absolute channel
            v2f a, bm;
            a.x = wok ? plane[cl * WW + w2i]       : 0.0f;   // A[m][k]
            a.y = wok ? plane[(cl + 1) * WW + w2i] : 0.0f;
            bm.x = (mn < NCH) ? wbase[cg * NCH + mn]       : 0.0f; // B[k][n]
            bm.y = (mn < NCH) ? wbase[(cg + 1) * NCH + mn] : 0.0f;
            acc = __builtin_amdgcn_wmma_f32_16x16x4_f32(false, a, false, bm,
                                                        (short)0, acc, false, false);
          }
        }
      }
    }
  }

  // C layout: VGPR r -> M = r + 8*half, N = lane&15
  if (mn < NCH) {
    const float bias = (mn < NCLS) ? cls_b[mn] : off_b[mn - NCLS];
#pragma unroll
    for (int r = 0; r < 8; ++r) {
      const int m  = r + (half << 3);
      const int wp = w0 + m;
      const size_t l = ((size_t)dz * HH + hy) * WW + wp;
      const float v = acc[r] + bias;
      if (mn < NCLS) {
        logits[((size_t)b * LL + l) * NCLS + mn] = v;
      } else {
        const int ax = mn - NCLS;  // 0=d,1=h,2=w (anchor channel order)
        const float anc = (ax == 0 ? (dz + 0.5f) : ax == 1 ? (hy + 0.5f) : (wp + 0.5f)) * 2.0f;
        centers[((size_t)b * LL + l) * 3 + ax] = anc + v;
      }
    }
  }
}

// ---------------------------------------------------------------------------
// K2: streaming top-13 of align = sigmoid(logit[cls])*iou^6 per (b, gt).
// ---------------------------------------------------------------------------
__global__ __launch_bounds__(256) void topk_kernel(
    const float* __restrict__ logits, const float* __restrict__ centers,
    const float* __restrict__ labels, float* __restrict__ tkval, int* __restrict__ tkidx)
{
  const int b = blockIdx.x / NMAX;
  const int n = blockIdx.x % NMAX;
  const int tid = threadIdx.x;
  const float* gl = labels + ((size_t)b * NMAX + n) * 5;
  float* ov = tkval + ((size_t)b * NMAX + n) * TOPKK;
  int*   oi = tkidx + ((size_t)b * NMAX + n) * TOPKK;
  const float clsf = gl[3];
  if (clsf == -100.0f) {                       // invalid gt -> no candidates
    if (tid < TOPKK) { ov[tid] = 0.0f; oi[tid] = -1; }
    return;
  }
  const float gx = gl[0], gy = gl[1], gz = gl[2], sg = gl[4];
  const float inv2 = 1.0f / (2.0f * sg * sg);
  const int lab = (int)clsf;

  float bv[TOPKK]; int bi[TOPKK];
#pragma unroll
  for (int k = 0; k < TOPKK; ++k) { bv[k] = -1.0f; bi[k] = -1; }

  for (int l = tid; l < LL; l += 256) {
    const float* pc = centers + ((size_t)b * LL + l) * 3;
    const float dx = gx - pc[0], dy = gy - pc[1], dzv = gz - pc[2];
    const float iou = expf(-(dx * dx + dy * dy + dzv * dzv) * inv2);
    const float i2 = iou * iou;
    const float x = logits[((size_t)b * LL + l) * NCLS + lab];
    const float align = sigmoidf_(x) * i2 * i2 * i2;
    if (align > bv[TOPKK - 1]) {
      int j = TOPKK - 1;
      while (j > 0 && align > bv[j - 1]) { bv[j] = bv[j - 1]; bi[j] = bi[j - 1]; --j; }
      bv[j] = align; bi[j] = l;
    }
  }

  __shared__ float sv[256 * TOPKK];
  __shared__ int   si[256 * TOPKK];
  for (int k = 0; k < TOPKK; ++k) { sv[tid * TOPKK + k] = bv[k]; si[tid * TOPKK + k] = bi[k]; }
  __syncthreads();
  if (tid == 0) {                              // merge: value desc, index asc
    for (int k = 0; k < TOPKK; ++k) {
      float best = -2.0f; int besti = 0x7fffffff; int bslot = -1;
      for (int s = 0; s < 256 * TOPKK; ++s) {
        const int idx = si[s];
        if (idx < 0) continue;
        const float v = sv[s];
        if (v > best || (v == best && idx < besti)) { best = v; besti = idx; bslot = s; }
      }
      if (bslot >= 0) { si[bslot] = -1; ov[k] = best; oi[k] = besti; }
      else            { ov[k] = 0.0f; oi[k] = -1; }
    }
  }
}

// ---------------------------------------------------------------------------
// K3: sparse TAL assignment over <=416 candidates per batch. Each column gets
// at most one gt after the max-IoU override; accumulate reg loss, cls-loss
// correction (fg term vs background term), and normalizer via atomics.
// ---------------------------------------------------------------------------
__global__ __launch_bounds__(448) void assign_kernel(
    const float* __restrict__ logits, const float* __restrict__ centers,
    const float* __restrict__ labels, const float* __restrict__ tkval,
    const int* __restrict__ tkidx, float* __restrict__ acc)
{
  const int b = blockIdx.x;
  const int tid = threadIdx.x;
  __shared__ int   sl[NSLOT];
  __shared__ float sv[NSLOT];
  __shared__ int   cn[NSLOT];
  __shared__ float ca[NSLOT], ci[NSLOT];
  __shared__ float gcx[NMAX], gcy[NMAX], gcz[NMAX], ginv[NMAX];
  __shared__ int   glab[NMAX];
  __shared__ float maxA[NMAX], maxI[NMAX];

  if (tid < NSLOT) {
    sl[tid] = tkidx[(size_t)b * NSLOT + tid];
    sv[tid] = tkval[(size_t)b * NSLOT + tid];
    cn[tid] = -1;
  }
  if (tid < NMAX) {
    const float* gl = labels + ((size_t)b * NMAX + tid) * 5;
    gcx[tid] = gl[0]; gcy[tid] = gl[1]; gcz[tid] = gl[2];
    const float sg = gl[4];
    ginv[tid] = 1.0f / (2.0f * sg * sg);
    const float cf = gl[3];
    glab[tid] = (cf == -100.0f) ? 0 : (int)cf;
  }
  __syncthreads();

  if (tid < NSLOT) {
    const int l = sl[tid];
    if (l >= 0 && sv[tid] > EPSF) {
      bool owner = true; int cnt = 0;
      for (int s = 0; s < NSLOT; ++s) {
        if (sl[s] == l && sv[s] > EPSF) { ++cnt; if (s < tid) owner = false; }
      }
      if (owner) {
        const float* pc = centers + ((size_t)b * LL + l) * 3;
        const float px = pc[0], py = pc[1], pz = pc[2];
        int na;
        if (cnt == 1) {
          na = tid / TOPKK;
        } else {                                 // pos_sum>1 -> argmax_n iou
          float best = -1.0f; na = 0;
          for (int nn = 0; nn < NMAX; ++nn) {
            const float dx = gcx[nn] - px, dy = gcy[nn] - py, dzv = gcz[nn] - pz;
            const float iou = expf(-(dx * dx + dy * dy + dzv * dzv) * ginv[nn]);
            if (iou > best) { best = iou; na = nn; }
          }
        }
        const float dx = gcx[na] - px, dy = gcy[na] - py, dzv = gcz[na] - pz;
        const float iou = expf(-(dx * dx + dy * dy + dzv * dzv) * ginv[na]);
        const float x = logits[((size_t)b * LL + l) * NCLS + glab[na]];
        const float i2 = iou * iou;
        ca[tid] = sigmoidf_(x) * i2 * i2 * i2;
        ci[tid] = iou;
        cn[tid] = na;
      }
    }
  }
  __syncthreads();
  if (tid == 0) {
    for (int nn = 0; nn < NMAX; ++nn) { maxA[nn] = 0.0f; maxI[nn] = 0.0f; }
    for (int s = 0; s < NSLOT; ++s) {
      if (cn[s] >= 0) {
        maxA[cn[s]] = fmaxf(maxA[cn[s]], ca[s]);
        maxI[cn[s]] = fmaxf(maxI[cn[s]], ci[s]);
      }
    }
  }
  __syncthreads();
  if (tid < NSLOT && cn[tid] >= 0) {
    const int na = cn[tid], l = sl[tid];
    const float score = ca[tid] * maxI[na] / (maxA[na] + EPSF);
    const float x = logits[((size_t)b * LL + l) * NCLS + glab[na]];
    const float p = sigmoidf_(x);
    const float sp = softplusf_(x), sn = softplusf_(-x);
    // fg term (weight=score, bce = score*sp(-x)+(1-score)*sp(x)) minus the
    // background term 0.75*p^2*sp(x) that K4 adds for every logit.
    const float corr = score * (score * sn + (1.0f - score) * sp) - 0.75f * p * p * sp;
    const float reg  = (1.0f - ci[tid]) * score;
    atomicAdd(acc + 1, corr);
    atomicAdd(acc + 2, reg);
    atomicAdd(acc + 3, score);
  }
}

// ---------------------------------------------------------------------------
// K4: background focal term over all B*L*5 logits.
// ---------------------------------------------------------------------------
__global__ __launch_bounds__(256) void bg_kernel(
    const float* __restrict__ logits, float* __restrict__ acc, int total)
{
  __shared__ float red[256];
  const int tid = threadIdx.x;
  const long i = (long)blockIdx.x * 256 + tid;
  float v = 0.0f;
  if (i < total) {
    const float x = logits[i];
    const float p = sigmoidf_(x);
    v = 0.75f * p * p * softplusf_(x);
  }
  red[tid] = v; __syncthreads();
  for (int s = 128; s > 0; s >>= 1) { if (tid < s) red[tid] += red[tid + s]; __syncthreads(); }
  if (tid == 0) atomicAdd(acc + 0, red[0]);
}

__global__ void init_kernel(float* __restrict__ acc) {
  if (threadIdx.x < 8) acc[threadIdx.x] = 0.0f;
}

__global__ void finalize_kernel(const float* __restrict__ acc, float* __restrict__ out) {
  out[0] = (acc[0] + acc[1] + acc[2]) / acc[3];
}

// ---------------------------------------------------------------------------
extern "C" void kernel_launch(void* const* d_in, const int* in_sizes, int n_in,
                              void* d_out, int out_size, void* d_ws, size_t ws_size,
                              hipStream_t stream)
{
  (void)in_sizes; (void)n_in; (void)out_size; (void)ws_size;
  const float* feat   = (const float*)d_in[0];
  const float* cls_w  = (const float*)d_in[1];
  const float* cls_b  = (const float*)d_in[2];
  const float* off_w  = (const float*)d_in[3];
  const float* off_b  = (const float*)d_in[4];
  const float* labels = (const float*)d_in[5];
  float* out = (float*)d_out;

  float* wsf    = (float*)d_ws;
  float* acc    = wsf;                                    // 8 floats
  float* logits = wsf + 16;                               // B*L*5
  float* centers = logits + (size_t)BATCH * LL * NCLS;    // B*L*3
  float* tkval   = centers + (size_t)BATCH * LL * 3;      // B*32*13
  int*   tkidx   = (int*)(tkval + (size_t)BATCH * NMAX * TOPKK);

  init_kernel<<<1, 32, 0, stream>>>(acc);
  conv_wmma_kernel<<<(BATCH * DD * HH * (WW / 16)) / 4, 128, 0, stream>>>(
      feat, cls_w, cls_b, off_w, off_b, logits, centers);
  topk_kernel<<<BATCH * NMAX, 256, 0, stream>>>(logits, centers, labels, tkval, tkidx);
  assign_kernel<<<BATCH, 448, 0, stream>>>(logits, centers, labels, tkval, tkidx, acc);
  const int total = BATCH * LL * NCLS;
  bg_kernel<<<(total + 255) / 256, 256, 0, stream>>>(logits, acc, total);
  finalize_kernel<<<1, 1, 0, stream>>>(acc, out);
}